// LightGCN_53944789237890
// MI455X (gfx1250) — compile-verified
//
#include <hip/hip_runtime.h>

#define N_USERS 100000
#define N_ITEMS 50000
#define N_NODES 150000
#define N_INTER 3200000
#define N_EDGES (2 * N_INTER)
#define DIM 64
#define N_QUERY 4096

typedef __attribute__((ext_vector_type(2))) float v2f;
typedef __attribute__((ext_vector_type(8))) float v8f;

// Guaranteed single-instruction hardware f32 atomic add at device scope
// (L2 atomic units), no CAS-loop fallback. Non-returning form -> STOREcnt,
// implicit s_wait_idle before S_ENDPGM guarantees completion at kernel end.
__device__ __forceinline__ void atomic_add_f32_dev(float* p, float v) {
  asm volatile("global_atomic_add_f32 %0, %1, off scope:SCOPE_DEV"
               :
               : "v"(p), "v"(v)
               : "memory");
}

// acc = emb = concat(user_emb, item_emb), vectorized as float4.
__global__ void lgcn_init(const float4* __restrict__ ue,
                          const float4* __restrict__ ie,
                          float4* __restrict__ acc,
                          float4* __restrict__ emb) {
  int i = blockIdx.x * blockDim.x + threadIdx.x;
  const int userVec = N_USERS * (DIM / 4);
  const int total   = N_NODES * (DIM / 4);
  if (i >= total) return;
  float4 v = (i < userVec) ? ue[i] : ie[i - userVec];
  acc[i] = v;
  emb[i] = v;
}

// One wave per edge: y[src] += val * x[dst]. 32 lanes cover 64 dims in 2 steps.
// Edge arrays are streamed with non-temporal loads so the L2 keeps the
// embedding tables resident; gathers/atomics stay coalesced and L2-local.
__global__ void lgcn_spmm(const int* __restrict__ esrc,
                          const int* __restrict__ edst,
                          const float* __restrict__ eval,
                          const float* __restrict__ x,
                          float* __restrict__ y) {
  int wave = (blockIdx.x * blockDim.x + threadIdx.x) >> 5;
  int lane = threadIdx.x & 31;
  if (wave >= N_EDGES) return;
  int   s = __builtin_nontemporal_load(esrc + wave);
  int   d = __builtin_nontemporal_load(edst + wave);
  float v = __builtin_nontemporal_load(eval + wave);
  const float* xr = x + (size_t)d * DIM;
  float*       yr = y + (size_t)s * DIM;
  float x0 = xr[lane];
  float x1 = xr[lane + 32];
  atomic_add_f32_dev(yr + lane,      v * x0);
  atomic_add_f32_dev(yr + lane + 32, v * x1);
}

// acc += layer_output
__global__ void lgcn_acc(float4* __restrict__ acc, const float4* __restrict__ e) {
  int i = blockIdx.x * blockDim.x + threadIdx.x;
  if (i >= N_NODES * (DIM / 4)) return;
  float4 a = acc[i];
  float4 b = e[i];
  a.x += b.x; a.y += b.y; a.z += b.z; a.w += b.w;
  acc[i] = a;
}

// gamma[j] = dot(acc[users[j]], acc[N_USERS+items[j]]) / 16
// One wave computes a 16-query tile as diag(U x V^T) via V_WMMA_F32_16X16X4_F32,
// accumulating K=64 as 16 chunks of K=4. Exact f32 math.
// A layout (16x4 f32): lanes 0-15 hold K=kb+0 (v0), kb+1 (v1); lanes 16-31 hold
// K=kb+2, kb+3. B (4x16) mirrors with lane = column N. D: VGPR g, lanes 0-15 ->
// M=g, N=lane; lanes 16-31 -> M=g+8, N=lane-16.
__global__ void lgcn_dot_wmma(const float* __restrict__ acc,
                              const int* __restrict__ users,
                              const int* __restrict__ items,
                              float* __restrict__ out) {
  int lane = threadIdx.x & 31;
  int tile = (blockIdx.x * blockDim.x + threadIdx.x) >> 5;
  int j0   = tile * 16;
  int r    = lane & 15;
  int half = lane >> 4;

  const float* up = acc + (size_t)users[j0 + r] * DIM;
  const float* vp = acc + (size_t)(N_USERS + items[j0 + r]) * DIM;

  v8f c = {};
#pragma unroll
  for (int kc = 0; kc < 16; ++kc) {
    int kb = kc * 4 + half * 2;
    v2f a; a.x = up[kb]; a.y = up[kb + 1];
    v2f b; b.x = vp[kb]; b.y = vp[kb + 1];
    c = __builtin_amdgcn_wmma_f32_16x16x4_f32(
        /*neg_a=*/false, a, /*neg_b=*/false, b,
        /*c_mod=*/(short)0, c, /*reuse_a=*/false, /*reuse_b=*/false);
  }

  // Diagonal M==N extraction, scaled by (1/4)^2 from light_out = acc/4.
  if (half == 0) {
    if (r < 8) out[j0 + r] = c[r] * 0.0625f;
  } else {
    if (r >= 8) out[j0 + r] = c[r - 8] * 0.0625f;
  }
}

extern "C" void kernel_launch(void* const* d_in, const int* in_sizes, int n_in,
                              void* d_out, int out_size, void* d_ws, size_t ws_size,
                              hipStream_t stream) {
  const float* ue    = (const float*)d_in[0];
  const float* ie    = (const float*)d_in[1];
  const int*   esrc  = (const int*)d_in[2];
  const int*   edst  = (const int*)d_in[3];
  const float* eval  = (const float*)d_in[4];
  const int*   users = (const int*)d_in[5];
  const int*   items = (const int*)d_in[6];
  float*       out   = (float*)d_out;

  const size_t embBytes = (size_t)N_NODES * DIM * sizeof(float);
  char* ws = (char*)d_ws;
  float* acc  = (float*)(ws);
  float* embA = (float*)(ws + embBytes);
  float* embB = (float*)(ws + 2 * embBytes);

  const int blk = 256;
  {
    int total = N_NODES * (DIM / 4);
    lgcn_init<<<(total + blk - 1) / blk, blk, 0, stream>>>(
        (const float4*)ue, (const float4*)ie, (float4*)acc, (float4*)embA);
  }

  float* cur = embA;
  float* nxt = embB;
  for (int layer = 0; layer < 3; ++layer) {
    hipMemsetAsync(nxt, 0, embBytes, stream);
    {
      long long threads = (long long)N_EDGES * 32;
      lgcn_spmm<<<(unsigned)((threads + blk - 1) / blk), blk, 0, stream>>>(
          esrc, edst, eval, cur, nxt);
    }
    {
      int total = N_NODES * (DIM / 4);
      lgcn_acc<<<(total + blk - 1) / blk, blk, 0, stream>>>(
          (float4*)acc, (const float4*)nxt);
    }
    float* t = cur; cur = nxt; nxt = t;
  }

  // 256 tiles of 16 queries, 1 wave each: 32 blocks x 256 threads, full waves
  // (EXEC all ones as WMMA requires).
  lgcn_dot_wmma<<<(N_QUERY / 16) * 32 / blk, blk, 0, stream>>>(acc, users, items, out);
}